// H2HGCN_13718125543774
// MI455X (gfx1250) — compile-verified
//
#include <hip/hip_runtime.h>
#include <hip/hip_bf16.h>
#include <math.h>

typedef float v2f __attribute__((ext_vector_type(2)));
typedef float v8f __attribute__((ext_vector_type(8)));

#define NN    30000
#define DEGN  16
#define DIMN  512
#define LDIMN 511
#define LAYERS 2
#define EPSF  1e-8f

__device__ __forceinline__ float selu_f(float x) {
    const float alpha = 1.6732632423543772f;
    const float scale = 1.0507009873554805f;
    return scale * (x > 0.0f ? x : alpha * (expf(x) - 1.0f));
}

__device__ __forceinline__ float wred(float v) {
#pragma unroll
    for (int m = 16; m >= 1; m >>= 1) v += __shfl_xor(v, m, 32);
    return v;
}

// WT[k][n] = W[n][k]  (512x512)
__global__ __launch_bounds__(256) void k_transpose(const float* __restrict__ W,
                                                   float* __restrict__ WT) {
    __shared__ float tile[16][17];
    const int bx = blockIdx.x * 16, by = blockIdx.y * 16;
    const int tx = threadIdx.x & 15, ty = threadIdx.x >> 4;
    tile[ty][tx] = W[(by + ty) * DIMN + bx + tx];
    __syncthreads();
    WT[(bx + ty) * DIMN + by + tx] = tile[tx][ty];
}

// LW = blockdiag(1, M)  with M 511x511, LW 512x512
__global__ __launch_bounds__(256) void k_build_lw(const float* __restrict__ M,
                                                  float* __restrict__ LW) {
    const int idx = blockIdx.x * 256 + threadIdx.x;
    const int i = idx >> 9, j = idx & 511;
    float v;
    if (i == 0 || j == 0) v = (i == 0 && j == 0) ? 1.0f : 0.0f;
    else                  v = M[(i - 1) * LDIMN + (j - 1)];
    LW[idx] = v;
}

// C[M x 512] = A[M x 512] @ B[512 x 512] (+bias, selu). One block = 16-row strip.
// 8 waves per block, wave w -> columns [64w, 64w+64), 4 f32 WMMA accumulators each.
__global__ __launch_bounds__(256) void k_gemm_wmma(const float* __restrict__ A,
                                                   const float* __restrict__ B,
                                                   float* __restrict__ C,
                                                   const float* __restrict__ bias,
                                                   int fuse_selu) {
    __shared__ float As[16 * DIMN];
    const int tid  = threadIdx.x;
    const int wv   = tid >> 5;
    const int lane = tid & 31;
    const int hl   = lane >> 4;   // half-wave select (K pairs)
    const int lm   = lane & 15;
    const long mBase = (long)blockIdx.x * 16;

    // Cooperative stage of the 16x512 A strip into LDS (32KB), 8 float4 per thread.
    const float4* A4 = (const float4*)(A + mBase * DIMN);
    float4* As4 = (float4*)As;
#pragma unroll
    for (int i = 0; i < 8; ++i) As4[tid + 256 * i] = A4[tid + 256 * i];
    __syncthreads();

    v8f acc[4];
    const v8f zero = {0.f, 0.f, 0.f, 0.f, 0.f, 0.f, 0.f, 0.f};
#pragma unroll
    for (int t = 0; t < 4; ++t) acc[t] = zero;

    const int nBase = wv * 64;
    for (int k = 0; k < DIMN; k += 4) {
        // A fragment 16x4 f32: lanes 0-15 hold K=k,k+1 ; lanes 16-31 hold K=k+2,k+3
        v2f a;
        a.x = As[lm * DIMN + k + 2 * hl];
        a.y = As[lm * DIMN + k + 2 * hl + 1];
        const float* Bk = B + (long)(k + 2 * hl) * DIMN;
#pragma unroll
        for (int t = 0; t < 4; ++t) {
            const int n = nBase + t * 16 + lm;
            v2f b;
            b.x = Bk[n];          // row k+2*hl
            b.y = Bk[DIMN + n];   // row k+2*hl+1
            acc[t] = __builtin_amdgcn_wmma_f32_16x16x4_f32(
                false, a, false, b, (short)0, acc[t], false, false);
        }
    }

    // D layout: VGPR r -> row r (lanes 0-15) / row r+8 (lanes 16-31), col = lane%16
#pragma unroll
    for (int t = 0; t < 4; ++t) {
        const int col = nBase + t * 16 + lm;
        const float bv = bias ? bias[col] : 0.0f;
#pragma unroll
        for (int r = 0; r < 8; ++r) {
            const long row = mBase + r + 8 * hl;
            float v = acc[t][r] + bv;
            if (fuse_selu) v = selu_f(v);
            C[row * DIMN + col] = v;
        }
    }
}

// In-place exp_map_zero + lorentz_normalize. One wave per node row.
__global__ __launch_bounds__(256) void k_expmap(float* __restrict__ H) {
    const int wv = threadIdx.x >> 5, lane = threadIdx.x & 31;
    const long node = (long)blockIdx.x * 8 + wv;
    float* row = H + node * DIMN;
    float vals[16];
    float s = 0.f;
#pragma unroll
    for (int i = 0; i < 16; ++i) {
        const int c = lane + 32 * i;
        const float v = row[c];
        vals[i] = v;
        if (c > 0) s += v * v;
    }
    s = wred(s);                       // ldv = sum(tail^2)
    const float nd = sqrtf(s + EPSF);
    const float t  = fminf(nd, 1.0f);
    const float sc = sinhf(t) / nd;
    const float first = sqrtf(1.0f + sc * sc * s);
#pragma unroll
    for (int i = 0; i < 16; ++i) {
        const int c = lane + 32 * i;
        row[c] = (c == 0) ? first : sc * vals[i];
    }
}

// Fused: gather neighbors from msg, Klein-weighted mean, k2h,
// lorentz->poincare, selu, poincare->lorentz, lorentz_normalize.
// One wave per node.
__global__ __launch_bounds__(256) void k_aggregate(const float* __restrict__ msg,
                                                   const int* __restrict__ adj,
                                                   const float* __restrict__ wgt,
                                                   float* __restrict__ out) {
    const int wv = threadIdx.x >> 5, lane = threadIdx.x & 31;
    const long node = (long)blockIdx.x * 8 + wv;

    float acc[16];
#pragma unroll
    for (int i = 0; i < 16; ++i) acc[i] = 0.f;
    float den = 0.f;

    for (int d = 0; d < DEGN; ++d) {
        const int j = adj[node * DEGN + d];
        const float* row = msg + (long)j * DIMN;
        const float inv0 = 1.0f / row[0];
        float vals[16];
        float s = 0.f;
#pragma unroll
        for (int i = 0; i < 16; ++i) {
            const int c = lane + 32 * i;
            const float xv = row[c] * inv0;
            vals[i] = xv;
            if (c > 0) s += xv * xv;
        }
        s = wred(s);
        const float n2 = fminf(fmaxf(s, 0.0f), 0.9f);
        const float lamb = wgt[node * DEGN + d] * rsqrtf(1.0f - n2);
        den += lamb;
#pragma unroll
        for (int i = 0; i < 16; ++i) {
            const int c = lane + 32 * i;
            if (c > 0) acc[i] += lamb * vals[i];
        }
    }

    const float invden = 1.0f / den;
    float km[16];
    float s = 0.f;
#pragma unroll
    for (int i = 0; i < 16; ++i) {
        const int c = lane + 32 * i;
        km[i] = (c > 0) ? acc[i] * invden : 0.f;
        s += km[i] * km[i];
    }
    s = wred(s);                             // |k_mean|^2
    const float n2m  = fminf(s, 0.9f);
    const float invs = rsqrtf(1.0f - n2m);   // h0 after k2h
    const float pf   = invs / (invs + 1.0f); // tail*invs / (invs+1)

    float sp[16];
    float s2 = 0.f;
#pragma unroll
    for (int i = 0; i < 16; ++i) {
        const int c = lane + 32 * i;
        if (c > 0) {
            const float sv = selu_f(km[i] * pf);
            sp[i] = sv;
            s2 += sv * sv;
        } else sp[i] = 0.f;
    }
    s2 = wred(s2);                           // |selu(p)|^2
    const float denom = 1.0f - s2 + EPSF;
    const float first = sqrtf(1.0f + 4.0f * s2 / (denom * denom));
#pragma unroll
    for (int i = 0; i < 16; ++i) {
        const int c = lane + 32 * i;
        out[node * DIMN + c] = (c == 0) ? first : 2.0f * sp[i] / denom;
    }
}

extern "C" void kernel_launch(void* const* d_in, const int* in_sizes, int n_in,
                              void* d_out, int out_size, void* d_ws, size_t ws_size,
                              hipStream_t stream) {
    const float* node_repr   = (const float*)d_in[0];
    const int*   adj         = (const int*)d_in[1];
    const float* weight      = (const float*)d_in[2];
    const float* lin_w       = (const float*)d_in[3];
    const float* lin_b       = (const float*)d_in[4];
    const float* msg_weights = (const float*)d_in[5];
    float* out = (float*)d_out;

    char* ws = (char*)d_ws;
    float* X  = (float*)ws;                                      // h   : N x 512
    float* Y  = (float*)(ws + (size_t)NN * DIMN * sizeof(float)); // msg : N x 512
    float* Bm = (float*)(ws + 2ull * NN * DIMN * sizeof(float));  // 512 x 512 operand

    // h0 = expmap(selu(node_repr @ lin_w^T + lin_b))
    dim3 tgrid(32, 32);
    k_transpose<<<tgrid, 256, 0, stream>>>(lin_w, Bm);
    k_gemm_wmma<<<NN / 16, 256, 0, stream>>>(node_repr, Bm, X, lin_b, 1);
    k_expmap<<<NN / 8, 256, 0, stream>>>(X);

    for (int step = 0; step < LAYERS; ++step) {
        k_build_lw<<<(DIMN * DIMN) / 256, 256, 0, stream>>>(
            msg_weights + (size_t)step * LDIMN * LDIMN, Bm);
        k_gemm_wmma<<<NN / 16, 256, 0, stream>>>(X, Bm, Y, nullptr, 0);
        k_aggregate<<<NN / 8, 256, 0, stream>>>(
            Y, adj, weight, (step == LAYERS - 1) ? out : X);
    }
}